// VAEWithVIStructure_84430467105115
// MI455X (gfx1250) — compile-verified
//
#include <hip/hip_runtime.h>
#include <hip/hip_bf16.h>

// out[i,d] = mu[i,d] + sigma[i,d] * scale(i) * e[i,d]
// scale(i) = sqrt(EPS) if parent[i] == -1 else 1.0   (see analysis: L is diagonal)
// B=2048, D=32 -> 65536 f32 elements = 16384 float4 lanes.

static constexpr float kSqrtEps = 0.031622776601683794f; // sqrt(0.001)

__global__ __launch_bounds__(256) void vae_vi_elementwise_async(
    const float* __restrict__ mu,
    const float* __restrict__ sigma,
    const float* __restrict__ e,
    const int*   __restrict__ parent,
    float*       __restrict__ out,
    int n4) // number of float4 elements
{
    __shared__ float4 s_mu[256];
    __shared__ float4 s_sg[256];
    __shared__ float4 s_e [256];

    const int t = blockIdx.x * 256 + threadIdx.x;
    if (t >= n4) return;

    const float4* gmu = reinterpret_cast<const float4*>(mu)    + t;
    const float4* gsg = reinterpret_cast<const float4*>(sigma) + t;
    const float4* ge  = reinterpret_cast<const float4*>(e)     + t;

    // LDS byte address = low 32 bits of the generic address of a __shared__
    // object (flat LDS aperture: the LDS offset is the address's low 32 bits, ISA 10.2).
    unsigned l_mu = (unsigned)(uintptr_t)&s_mu[threadIdx.x];
    unsigned l_sg = (unsigned)(uintptr_t)&s_sg[threadIdx.x];
    unsigned l_e  = (unsigned)(uintptr_t)&s_e [threadIdx.x];

    // gfx1250 async VMEM->LDS (tracked by ASYNCcnt), GV addressing mode:
    //   global_load_async_to_lds_b128 vdst(lds-addr), vaddr(64b), off
    asm volatile("global_load_async_to_lds_b128 %0, %1, off"
                 :: "v"(l_mu), "v"(gmu) : "memory");
    asm volatile("global_load_async_to_lds_b128 %0, %1, off"
                 :: "v"(l_sg), "v"(gsg) : "memory");
    asm volatile("global_load_async_to_lds_b128 %0, %1, off"
                 :: "v"(l_e),  "v"(ge)  : "memory");

    // Overlap the (scalar-per-row) parent fetch with the async copies.
    const int row = t >> 3;                       // 8 float4 per 32-wide row
    const float scale = (parent[row] == -1) ? kSqrtEps : 1.0f;

    asm volatile("s_wait_asynccnt 0x0" ::: "memory");

    const float4 m = s_mu[threadIdx.x];
    const float4 s = s_sg[threadIdx.x];
    const float4 v = s_e [threadIdx.x];

    float4 r;
    r.x = fmaf(s.x, scale * v.x, m.x);
    r.y = fmaf(s.y, scale * v.y, m.y);
    r.z = fmaf(s.z, scale * v.z, m.z);
    r.w = fmaf(s.w, scale * v.w, m.w);

    reinterpret_cast<float4*>(out)[t] = r;
}

extern "C" void kernel_launch(void* const* d_in, const int* in_sizes, int n_in,
                              void* d_out, int out_size, void* d_ws, size_t ws_size,
                              hipStream_t stream) {
    // setup_inputs order: mu, sigma, gamma (dead in the math), e, parent
    const float* mu     = (const float*)d_in[0];
    const float* sigma  = (const float*)d_in[1];
    const float* e      = (const float*)d_in[3];
    const int*   parent = (const int*)d_in[4];
    float* out = (float*)d_out;

    const int n  = out_size;          // 2048*32 = 65536 floats
    const int n4 = n / 4;             // 16384 float4 lanes
    const int block = 256;
    const int grid  = (n4 + block - 1) / block;   // 64 blocks

    vae_vi_elementwise_async<<<grid, block, 0, stream>>>(mu, sigma, e, parent, out, n4);
}